// MLPPredictor_5609227288773
// MI455X (gfx1250) — compile-verified
//
#include <hip/hip_runtime.h>

// ---------------------------------------------------------------------------
// Edge MLP: score = relu((h[src]*h[dst]) @ W1 + b1) @ W2 + b2
//   h: [50000,128] f32, W1: [128,64], b1: [64], W2: [64,2], b2: [2]
//   src/dst: [600000] int32, out: [600000,2] f32
//
// Strategy (MI455X / gfx1250, wave32):
//  * h (25.6 MB) fits in the 192 MB L2 -> random row gathers are L2 hits.
//  * Main GEMM in full fp32 via V_WMMA_F32_16X16X4_F32 (exact fp32 math).
//  * One wave = one 16-edge tile: coalesced float4 row gathers -> padded LDS
//    e-tile -> 128 f32 WMMAs (32 K-chunks x 4 N-tiles of 16).
//  * W1 pre-repacked (prologue kernel into d_ws) into per-lane B-fragment
//    order; the compiler hoists all 128 B fragments into registers, and we
//    launch a device-sized grid-stride loop so each wave amortizes that
//    preload over ~6 tiles (register-resident W1, zero B traffic per tile).
// ---------------------------------------------------------------------------

#define D_IN        128
#define D_HID       64
#define OUT_CLS     2
#define TILE_M      16
#define WAVES_PB    4
#define BLOCK_THR   (WAVES_PB * 32)
#define GRID_BLOCKS 1536  // device-filling grid; waves grid-stride over tiles
#define E_STRIDE    132   // padded e-tile row stride (floats); 132*4B keeps 16B
                          // alignment and gives conflict-free A-fragment reads
#define H_STRIDE    68    // padded hid-tile row stride (floats)

typedef __attribute__((ext_vector_type(2))) float v2f;
typedef __attribute__((ext_vector_type(8))) float v8f;

// ---------------------------------------------------------------------------
// Repack W1 [128][64] row-major into WMMA-B fragment order for
// V_WMMA_F32_16X16X4_F32.  B is 4x16 (KxN), 2 VGPRs / 32 lanes:
//   lane l (l<16):  VGPR0 = B[0][l],  VGPR1 = B[1][l]
//   lane l (l>=16): VGPR0 = B[2][l-16], VGPR1 = B[3][l-16]
// For K-chunk c (rows 4c..4c+3) and N-tile t (cols 16t..16t+15), lane l needs
//   { W1[4c + 2*(l>>4)][16t + (l&15)], W1[4c + 2*(l>>4) + 1][16t + (l&15)] }
// stored contiguously at Bp[((c*4 + t)*32 + l)*2 + {0,1}]  (32 KB total).
// ---------------------------------------------------------------------------
__global__ void repack_w1_kernel(const float* __restrict__ W1,
                                 float* __restrict__ Bp) {
  int g = blockIdx.x * blockDim.x + threadIdx.x;   // 0 .. 4095
  if (g >= 32 * 4 * 32) return;
  int l = g & 31;
  int t = (g >> 5) & 3;
  int c = g >> 7;
  int k = 4 * c + 2 * (l >> 4);
  int n = 16 * t + (l & 15);
  Bp[2 * g + 0] = W1[k * D_HID + n];
  Bp[2 * g + 1] = W1[(k + 1) * D_HID + n];
}

// ---------------------------------------------------------------------------
// Main fused kernel: one wave handles one 16-edge tile per grid-stride step.
// ---------------------------------------------------------------------------
__global__ __launch_bounds__(BLOCK_THR)
void edge_mlp_kernel(const float* __restrict__ h,
                     const float* __restrict__ Bp,   // repacked W1 fragments
                     const float* __restrict__ b1,
                     const float* __restrict__ W2,
                     const float* __restrict__ b2,
                     const int*   __restrict__ src,
                     const int*   __restrict__ dst,
                     float*       __restrict__ out,
                     int n_edges) {
  __shared__ float s_par[D_HID * OUT_CLS + D_HID + OUT_CLS];     // W2|b1|b2
  __shared__ float s_e[WAVES_PB][TILE_M * E_STRIDE];             // ~33 KB

  const int lane = threadIdx.x & 31;
  const int wave = threadIdx.x >> 5;
  const int m_l  = lane & 15;          // row (edge) / col (N) within tile
  const int k_hi = (lane >> 4) << 1;   // 0 or 2: K sub-pair owned by lane half
  const int r_hi = (lane >> 4) << 3;   // 0 or 8: C/D row offset for lane half

  // Stage W2 (64x2), b1 (64), b2 (2) into LDS once per block.
  for (int i = threadIdx.x; i < D_HID * OUT_CLS + D_HID + OUT_CLS;
       i += BLOCK_THR) {
    float v;
    if (i < D_HID * OUT_CLS)               v = W2[i];
    else if (i < D_HID * OUT_CLS + D_HID)  v = b1[i - D_HID * OUT_CLS];
    else                                   v = b2[i - D_HID * OUT_CLS - D_HID];
    s_par[i] = v;
  }
  __syncthreads();

  float* et = s_e[wave];

  const int n_tiles      = (n_edges + TILE_M - 1) / TILE_M;
  const int stride_tiles = gridDim.x * WAVES_PB;

  for (int tile = blockIdx.x * WAVES_PB + wave; tile < n_tiles;
       tile += stride_tiles) {
    const int e0 = tile * TILE_M;

    // Prefetch the next grid-stride tile's index data (speculative; silently
    // dropped if the address is past the end of the arrays).
    __builtin_prefetch(src + e0 + stride_tiles * TILE_M, 0, 0);
    __builtin_prefetch(dst + e0 + stride_tiles * TILE_M, 0, 0);

    // Per-lane edge indices (lanes 0..15 own one edge; high half duplicates).
    int ei = e0 + m_l;
    if (ei >= n_edges) ei = n_edges - 1;           // tail clamp (output guarded)
    const int my_s = src[ei];
    const int my_d = dst[ei];

    // ---- Stage e = h[src]*h[dst] tile into LDS: coalesced 512B row loads ----
    #pragma unroll 4
    for (int m = 0; m < TILE_M; ++m) {
      const int sm = __shfl(my_s, m, 32);
      const int dm = __shfl(my_d, m, 32);
      const float4 a = *(const float4*)(h + (size_t)sm * D_IN + lane * 4);
      const float4 b = *(const float4*)(h + (size_t)dm * D_IN + lane * 4);
      float4 p;
      p.x = a.x * b.x; p.y = a.y * b.y; p.z = a.z * b.z; p.w = a.w * b.w;
      *(float4*)(et + m * E_STRIDE + lane * 4) = p;
    }

    // ---- hid = e @ W1 via fp32 WMMA: K tiled by 4, N tiled by 16 (x4) ----
    v8f acc0 = {}, acc1 = {}, acc2 = {}, acc3 = {};
    #pragma unroll
    for (int c = 0; c < D_IN / 4; ++c) {
      // A fragment: lane l holds e[m_l][4c + k_hi .. +1] (conflict-free b64).
      const v2f av = *(const v2f*)(et + m_l * E_STRIDE + 4 * c + k_hi);
      // B fragments: loop-invariant, hoisted to registers by the compiler.
      const float* bp = Bp + ((size_t)(c * 4) * 32 + lane) * 2;
      const v2f bv0 = *(const v2f*)(bp);
      const v2f bv1 = *(const v2f*)(bp + 64);
      const v2f bv2 = *(const v2f*)(bp + 128);
      const v2f bv3 = *(const v2f*)(bp + 192);
      acc0 = __builtin_amdgcn_wmma_f32_16x16x4_f32(false, av, false, bv0,
                                                   (short)0, acc0, false, false);
      acc1 = __builtin_amdgcn_wmma_f32_16x16x4_f32(false, av, false, bv1,
                                                   (short)0, acc1, false, false);
      acc2 = __builtin_amdgcn_wmma_f32_16x16x4_f32(false, av, false, bv2,
                                                   (short)0, acc2, false, false);
      acc3 = __builtin_amdgcn_wmma_f32_16x16x4_f32(false, av, false, bv3,
                                                   (short)0, acc3, false, false);
    }

    // ---- bias + ReLU in C/D layout, write hid tile to LDS (reuse et) ----
    // C/D layout: VGPR r, lane l -> row (r + 8*(l>=16)), col (l&15).
    #define EPILOGUE(ACC, T)                                                 \
      {                                                                      \
        const float bias = s_par[D_HID * OUT_CLS + (T) * 16 + m_l];          \
        _Pragma("unroll") for (int r = 0; r < 8; ++r) {                      \
          float v = ACC[r] + bias;                                           \
          v = v > 0.0f ? v : 0.0f;                                           \
          et[(r + r_hi) * H_STRIDE + (T) * 16 + m_l] = v;                    \
        }                                                                    \
      }
    EPILOGUE(acc0, 0) EPILOGUE(acc1, 1) EPILOGUE(acc2, 2) EPILOGUE(acc3, 3)
    #undef EPILOGUE

    // ---- score = hid @ W2 + b2 : 32 outputs, lane = edge*2 + class ----
    const int m_out = lane >> 1;
    const int cl    = lane & 1;
    float sum = s_par[D_HID * OUT_CLS + D_HID + cl];       // b2[cl]
    #pragma unroll
    for (int k = 0; k < D_HID; ++k)
      sum += et[m_out * H_STRIDE + k] * s_par[k * OUT_CLS + cl];

    const int oi = e0 * OUT_CLS + lane;                    // coalesced stores
    if (oi < n_edges * OUT_CLS) out[oi] = sum;
  }
}

// ---------------------------------------------------------------------------
// Host-side launch
// ---------------------------------------------------------------------------
extern "C" void kernel_launch(void* const* d_in, const int* in_sizes, int n_in,
                              void* d_out, int out_size, void* d_ws,
                              size_t ws_size, hipStream_t stream) {
  (void)n_in; (void)out_size; (void)ws_size;
  const float* h   = (const float*)d_in[0];
  const float* W1  = (const float*)d_in[1];
  const float* b1  = (const float*)d_in[2];
  const float* W2  = (const float*)d_in[3];
  const float* b2  = (const float*)d_in[4];
  const int*   src = (const int*)d_in[5];
  const int*   dst = (const int*)d_in[6];
  float*       out = (float*)d_out;
  float*       Bp  = (float*)d_ws;          // 32 KB repacked-W1 scratch

  const int n_edges = in_sizes[5];

  // 1) Repack W1 into WMMA-B fragment order (stream-ordered before main).
  repack_w1_kernel<<<(32 * 4 * 32 + 255) / 256, 256, 0, stream>>>(W1, Bp);

  // 2) Fused gather + MLP: fixed device-filling grid, waves grid-stride over
  //    16-edge tiles so the register-resident W1 preload is amortized.
  const int n_tiles = (n_edges + TILE_M - 1) / TILE_M;
  int blocks = (n_tiles + WAVES_PB - 1) / WAVES_PB;
  if (blocks > GRID_BLOCKS) blocks = GRID_BLOCKS;
  edge_mlp_kernel<<<blocks, BLOCK_THR, 0, stream>>>(h, Bp, b1, W2, b2, src,
                                                    dst, out, n_edges);
}